// MaskPool_25434796327455
// MI455X (gfx1250) — compile-verified
//
#include <hip/hip_runtime.h>
#include <hip/hip_bf16.h>

// Masked spatial max-pool, MI455X (gfx1250, wave32).
//
// x:    (B=16, C=64, H=512, W=512) float32
// mask: (B=16, 1,    H=512, W=512) int32 (0/1)
// out:  (B=16, C=64) float32
//
// HBM-bound streaming reduction: ~1.07 GB of x (read once, non-temporal) +
// 16.8 MB of mask (L2-resident across channel re-reads; 192 MB L2) ->
// ~47 us floor at 23.3 TB/s. No matmul content -> WMMA inapplicable; the
// CDNA5 levers used are TH cache hints, b128 load clauses, wave32 shuffles.
//
// CG=2 channels per block: each mask int4 is loaded once into registers and
// applied to two x streams, halving mask L2 traffic vs one-channel blocks,
// while each block still covers the FULL spatial extent -> the per-batch
// mask count is computed exactly per block (no atomics, no init pass).

typedef float f32x4 __attribute__((ext_vector_type(4)));
typedef int   i32x4 __attribute__((ext_vector_type(4)));

#define MP_H        512
#define MP_W        512
#define MP_HW       (MP_H * MP_W)      // 262144
#define MP_C        64
#define MP_B        16
#define MP_CG       2                  // channels per block
#define MP_THREADS  256                // 8 wave32s per block

__global__ __launch_bounds__(MP_THREADS)
void MaskPool_25434796327455_kernel(const float* __restrict__ x,
                                    const int*   __restrict__ mask,
                                    float*       __restrict__ out)
{
    const int cg = blockIdx.x;         // 0..31  (channel group)
    const int b  = blockIdx.y;         // 0..15
    const int c0 = cg * MP_CG;         // first channel of this group

    const size_t xbase = ((size_t)b * MP_C + (size_t)c0) * (size_t)MP_HW;
    const size_t mbase = (size_t)b * (size_t)MP_HW;

    const f32x4* __restrict__ xv0 = (const f32x4*)(x + xbase);            // channel c0
    const f32x4* __restrict__ xv1 = (const f32x4*)(x + xbase + MP_HW);    // channel c0+1
    const i32x4* __restrict__ mv  = (const i32x4*)(mask + mbase);

    const float NEG_INF = -__builtin_inff();
    float best0 = NEG_INF;
    float best1 = NEG_INF;
    int   cnt   = 0;

    const int nvec = MP_HW / 4;        // 65536 float4/int4 per channel
    // 256 iterations/thread; unroll 8 -> ~24 outstanding b128 loads/thread.
    #pragma unroll 8
    for (int i = threadIdx.x; i < nvec; i += MP_THREADS) {
        i32x4 m  = mv[i];                                // RT: L2-resident, reused
        f32x4 v0 = __builtin_nontemporal_load(&xv0[i]);  // th:NT - streamed once
        f32x4 v1 = __builtin_nontemporal_load(&xv1[i]);  // th:NT - streamed once

        best0 = fmaxf(best0, (m.x != 0) ? v0.x : NEG_INF);
        best0 = fmaxf(best0, (m.y != 0) ? v0.y : NEG_INF);
        best0 = fmaxf(best0, (m.z != 0) ? v0.z : NEG_INF);
        best0 = fmaxf(best0, (m.w != 0) ? v0.w : NEG_INF);

        best1 = fmaxf(best1, (m.x != 0) ? v1.x : NEG_INF);
        best1 = fmaxf(best1, (m.y != 0) ? v1.y : NEG_INF);
        best1 = fmaxf(best1, (m.z != 0) ? v1.z : NEG_INF);
        best1 = fmaxf(best1, (m.w != 0) ? v1.w : NEG_INF);

        cnt += (m.x != 0) + (m.y != 0) + (m.z != 0) + (m.w != 0);
    }

    // ---- wave32 shuffle reduction (warpSize == 32 on gfx1250) ----
    #pragma unroll
    for (int off = 16; off > 0; off >>= 1) {
        best0 = fmaxf(best0, __shfl_xor(best0, off, 32));
        best1 = fmaxf(best1, __shfl_xor(best1, off, 32));
        cnt += __shfl_xor(cnt, off, 32);
    }

    // ---- cross-wave combine: 8 wave32s per block ----
    __shared__ float smax0[MP_THREADS / 32];
    __shared__ float smax1[MP_THREADS / 32];
    __shared__ int   scnt [MP_THREADS / 32];
    const int lane = threadIdx.x & 31;
    const int wave = threadIdx.x >> 5;
    if (lane == 0) {
        smax0[wave] = best0;
        smax1[wave] = best1;
        scnt [wave] = cnt;
    }
    __syncthreads();

    if (threadIdx.x == 0) {
        float m0 = smax0[0];
        float m1 = smax1[0];
        int   n  = scnt[0];
        #pragma unroll
        for (int w = 1; w < MP_THREADS / 32; ++w) {
            m0 = fmaxf(m0, smax0[w]);
            m1 = fmaxf(m1, smax1[w]);
            n += scnt[w];
        }
        // Block covers the FULL spatial extent -> n is the exact per-batch
        // mask count. Empty-mask batches produce 0 (reference semantics).
        float* o = out + (size_t)b * MP_C + (size_t)c0;
        o[0] = (n > 0) ? m0 : 0.0f;
        o[1] = (n > 0) ? m1 : 0.0f;
    }
}

extern "C" void kernel_launch(void* const* d_in, const int* in_sizes, int n_in,
                              void* d_out, int out_size, void* d_ws, size_t ws_size,
                              hipStream_t stream)
{
    (void)in_sizes; (void)n_in; (void)d_ws; (void)ws_size; (void)out_size;

    const float* x    = (const float*)d_in[0];  // (16,64,512,512) f32
    const int*   mask = (const int*)  d_in[1];  // (16,1,512,512) i32
    float*       out  = (float*)d_out;          // (16,64) f32

    dim3 grid(MP_C / MP_CG, MP_B);    // 32 x 16 = 512 blocks
    dim3 block(MP_THREADS);           // 8 wave32s
    MaskPool_25434796327455_kernel<<<grid, block, 0, stream>>>(x, mask, out);
}